// DEQ_27582279975269
// MI455X (gfx1250) — compile-verified
//
#include <hip/hip_runtime.h>

// DEQ fixed-point solve, fully fused, f32 WMMA (V_WMMA_F32_16X16X4_F32), gfx1250.
//
// Formulation (transposed so the WMMA A-operand is the loop-invariant weight):
//   D[state_out(16), row(16)] = A_tile[state_out(16), state_in(4)] x X[state_in(4), row(16)] + C
// 20 states -> 2 M-tiles (0..15, 16..19 zero-padded) x 5 K-chunks of 4.
// D-layout == next iteration's bias/accumulator layout; D -> B-operand needs only
// a lanes 0-15 <-> 16-31 swap (ds_swizzle SWAPX16).

typedef __attribute__((ext_vector_type(2))) float v2f;
typedef __attribute__((ext_vector_type(8))) float v8f;

#define NS 20
#define TOTAL_APPS 11   // total applications of step(); contraction ~0.12 => ~1e-9 rel residual

__device__ __forceinline__ float swap16(float x) {
  // group-of-32 swizzle: xor_mask=0x10, and_mask=0x1f -> swap lane halves. Full-EXEC only.
  return __int_as_float(__builtin_amdgcn_ds_swizzle(__float_as_int(x), 0x401F));
}

__device__ __forceinline__ float fast_tanh(float x) {
#if __has_builtin(__builtin_amdgcn_tanhf)
  return __builtin_amdgcn_tanhf(x);   // v_tanh_f32 (gfx1250 trans op)
#else
  return tanhf(x);
#endif
}

#define WMMA4(A, Bv, Cv) \
  __builtin_amdgcn_wmma_f32_16x16x4_f32(false, (A), false, (Bv), (short)0, (Cv), false, false)

__global__ __launch_bounds__(256) void deq_fp_kernel(
    const float* __restrict__ u,
    const float* __restrict__ Aw, const float* __restrict__ Ab,
    const float* __restrict__ Bw, const float* __restrict__ Bb,
    const float* __restrict__ Cw, const float* __restrict__ Cb,
    const float* __restrict__ Dw, const float* __restrict__ Db,
    float* __restrict__ y, int nrows)
{
  const int lane = threadIdx.x & 31;
  const int tile = blockIdx.x * (blockDim.x >> 5) + (threadIdx.x >> 5);
  const int r0 = tile * 16;
  if (r0 >= nrows) return;                 // wave-uniform exit keeps EXEC all-ones below
  const int  rl = lane & 15;
  const bool lo = lane < 16;
  const int  row = (r0 + rl < nrows) ? (r0 + rl) : (nrows - 1);
  const float uv = u[row];

  // ---- WMMA A-operand tiles of A_w (loop invariant). A 16x4 f32 layout:
  // vgpr j, lanes 0-15: (M=lane, K=4*kc+j); lanes 16-31: (M=lane-16, K=4*kc+2+j)
  v2f a[2][5];
  #pragma unroll
  for (int mt = 0; mt < 2; ++mt) {
    const int m  = 16 * mt + rl;
    const int mc = (m < NS) ? m : 0;
    const float mask = (m < NS) ? 1.f : 0.f;
    #pragma unroll
    for (int kc = 0; kc < 5; ++kc) {
      const int kb = 4 * kc + (lo ? 0 : 2);
      a[mt][kc].x = mask * Aw[mc * NS + kb];
      a[mt][kc].y = mask * Aw[mc * NS + kb + 1];
    }
  }

  // ---- bias c = A_b + B_b + B_w*u in WMMA C/D layout (state = M, row = N)
  // D0 covers states 0..15, D1 covers 16..31 (>=20 zero-padded)
  v8f c0, c1v;
  #pragma unroll
  for (int v = 0; v < 8; ++v) {
    const int s = (lo ? 0 : 8) + v;
    c0[v] = Ab[s] + Bb[s] + Bw[s] * uv;
  }
  #pragma unroll
  for (int v = 0; v < 8; ++v) {
    const int s  = (lo ? 16 : 24) + v;
    const int sc = (s < NS) ? s : 0;
    const float val = Ab[sc] + Bb[sc] + Bw[sc] * uv;
    c1v[v] = (s < NS) ? val : 0.f;
  }

  // d holds pre-activation; tanh(d) is the current iterate. d=c <=> x=0 start.
  v8f d0 = c0, d1 = c1v;

  for (int it = 0; it < TOTAL_APPS - 1; ++it) {
    // x = tanh(d), elementwise in D layout
    v8f t0;
    #pragma unroll
    for (int v = 0; v < 8; ++v) t0[v] = fast_tanh(d0[v]);
    const float t16 = fast_tanh(d1[0]);
    const float t17 = fast_tanh(d1[1]);
    const float t18 = fast_tanh(d1[2]);
    const float t19 = fast_tanh(d1[3]);

    // cross-half copies (hoisted: must run with full EXEC)
    const float s00 = swap16(t0[0]);
    const float s01 = swap16(t0[1]);
    const float s02 = swap16(t0[2]);
    const float s03 = swap16(t0[3]);
    const float s04 = swap16(t0[4]);
    const float s05 = swap16(t0[5]);
    const float s06 = swap16(t0[6]);
    const float s07 = swap16(t0[7]);
    const float s18 = swap16(t18);
    const float s19 = swap16(t19);

    // B-operand chunks: vgpr0 = K=4kc (lanes<16) / K=4kc+2 (lanes>=16); vgpr1 = +1
    v2f b0, b1, b2, b3, b4;
    b0.x = lo ? t0[0] : s02;  b0.y = lo ? t0[1] : s03;   // states 0..3
    b1.x = lo ? t0[4] : s06;  b1.y = lo ? t0[5] : s07;   // states 4..7
    b2.x = lo ? s00 : t0[2];  b2.y = lo ? s01 : t0[3];   // states 8..11
    b3.x = lo ? s04 : t0[6];  b3.y = lo ? s05 : t0[7];   // states 12..15
    b4.x = lo ? t16 : s18;    b4.y = lo ? t17 : s19;     // states 16..19

    // d = A_w * x + c  (accumulator chained through the 5 K-chunks)
    d0 = WMMA4(a[0][0], b0, c0);
    d0 = WMMA4(a[0][1], b1, d0);
    d0 = WMMA4(a[0][2], b2, d0);
    d0 = WMMA4(a[0][3], b3, d0);
    d0 = WMMA4(a[0][4], b4, d0);
    d1 = WMMA4(a[1][0], b0, c1v);
    d1 = WMMA4(a[1][1], b1, d1);
    d1 = WMMA4(a[1][2], b2, d1);
    d1 = WMMA4(a[1][3], b3, d1);
    d1 = WMMA4(a[1][4], b4, d1);
  }

  // x_star = tanh(d);  y = C x_star + C_b + D u + D_b
  v8f xs;
  #pragma unroll
  for (int v = 0; v < 8; ++v) xs[v] = fast_tanh(d0[v]);
  const float x16 = fast_tanh(d1[0]);
  const float x17 = fast_tanh(d1[1]);
  const float x18 = fast_tanh(d1[2]);
  const float x19 = fast_tanh(d1[3]);

  float p = 0.f;
  #pragma unroll
  for (int v = 0; v < 8; ++v) p += Cw[(lo ? 0 : 8) + v] * xs[v];
  const float w16 = lo ? 1.f : 0.f;   // states 16..19 live in lower lanes only
  p += w16 * (Cw[16] * x16 + Cw[17] * x17 + Cw[18] * x18 + Cw[19] * x19);

  const float ptot = p + swap16(p);   // full-EXEC reduction across halves
  if (lo) {
    const int r = r0 + rl;
    if (r < nrows) y[r] = ptot + Cb[0] + uv * Dw[0] + Db[0];
  }
}

extern "C" void kernel_launch(void* const* d_in, const int* in_sizes, int n_in,
                              void* d_out, int out_size, void* d_ws, size_t ws_size,
                              hipStream_t stream) {
  const float* u  = (const float*)d_in[0];
  const float* Aw = (const float*)d_in[1];
  const float* Ab = (const float*)d_in[2];
  const float* Bw = (const float*)d_in[3];
  const float* Bb = (const float*)d_in[4];
  const float* Cw = (const float*)d_in[5];
  const float* Cb = (const float*)d_in[6];
  const float* Dw = (const float*)d_in[7];
  const float* Db = (const float*)d_in[8];
  const int nrows = in_sizes[0];              // u is [B,1]
  const int rows_per_block = (256 / 32) * 16; // 8 waves x 16 rows = 128
  const int grid = (nrows + rows_per_block - 1) / rows_per_block;
  deq_fp_kernel<<<grid, 256, 0, stream>>>(u, Aw, Ab, Bw, Bb, Cw, Cb, Dw, Db,
                                          (float*)d_out, nrows);
}